// RippleNetPlus_54245436948483
// MI455X (gfx1250) — compile-verified
//
#include <hip/hip_runtime.h>
#include <math.h>

// RippleNetPlus forward for MI455X (gfx1250, wave32).
// B=1024, N_HOP=2, N_MEM=32, DIM=64.
// One workgroup (256 threads = 8 wave32) per batch element.
// HBM-bound on streaming Rs (1 GiB, ~46us floor at 23.3 TB/s).
//  - Rs streamed via double-buffered GLOBAL_LOAD_ASYNC_TO_LDS_B128 (ASYNCcnt)
//  - z @ W1^T GEMM (32x64x256 per sample) via V_WMMA_F32_16X16X4_F32

typedef float v8f __attribute__((ext_vector_type(8)));
typedef float v2f __attribute__((ext_vector_type(2)));
typedef int   v4i __attribute__((ext_vector_type(4)));

// async-to-LDS builtin pointer types: (int4 AS1*, int4 AS3*, Ii, Ii)
typedef __attribute__((address_space(1))) v4i gvec4i;
typedef __attribute__((address_space(3))) v4i lvec4i;

#define NMEM 32
#define DIM  64
#define RH_STRIDE 65          // pad: 64 banks
#define TILE_STRIDE 68        // padded row stride (floats) of staged R tile
#define TILE_F (DIM * TILE_STRIDE)

__device__ __forceinline__ float fast_tanhf(float x) {
    x = fmaxf(fminf(x, 15.0f), -15.0f);
    float t = __expf(2.0f * x);
    return (t - 1.0f) / (t + 1.0f);
}

__global__ __launch_bounds__(256) void ripplenet_fwd_kernel(
    const float* __restrict__ hs,      // (B,2,32,64)
    const float* __restrict__ Rs,      // (B,2,32,64,64)
    const float* __restrict__ ts,      // (B,2,32,64)
    const float* __restrict__ vs,      // (B,64)
    const float* __restrict__ W1_w,    // (64,256)
    const float* __restrict__ W1_b,    // (64)
    const float* __restrict__ W2_w,    // (1,64)
    const float* __restrict__ W2_b,    // (1)
    const float* __restrict__ Wmem_w,  // (2,64,192)
    const float* __restrict__ Wmem_b,  // (2,64)
    const float* __restrict__ z_w,     // (64,128)
    const float* __restrict__ z_b,     // (64)
    float* __restrict__ out)           // (B)
{
    const int b    = blockIdx.x;
    const int tid  = threadIdx.x;
    const int lane = tid & 31;
    const int wave = tid >> 5;

    __shared__ __align__(16) float Rbuf[2 * TILE_F]; // async R tile double-buffer
    __shared__ float h_s[NMEM * DIM];
    __shared__ float Rh_s[NMEM * RH_STRIDE];
    __shared__ float M_s[DIM];
    __shared__ float o_s[DIM];
    __shared__ float onew_s[DIM];
    __shared__ float Z_s[NMEM];
    __shared__ float g_s[NMEM];
    __shared__ float w1b_s[DIM];
    __shared__ float w2_s[DIM];
    __shared__ float red_s[DIM];

    // ---- init persistent state: M = o = vs[b] ; stage small weights ----
    if (tid < DIM) {
        float v = vs[(size_t)b * DIM + tid];
        M_s[tid]   = v;
        o_s[tid]   = v;
        w1b_s[tid] = W1_b[tid];
        w2_s[tid]  = W2_w[tid];
    }

    // WMMA tile assignment: 8 waves <-> 8 output tiles of the 32x64 GEMM
    const int tile_m = wave >> 2;            // 0..1  rows (m)
    const int tile_n = wave & 3;             // 0..3  cols (d)
    const int half   = lane >> 4;            // K sub-pair selector
    const int l16    = lane & 15;
    const int mrow   = tile_m * 16 + l16;    // A-row held by this lane
    const int ncol   = tile_n * 16 + l16;    // B-col held by this lane
    const float* W1row = W1_w + (size_t)ncol * 256;

    // z[m][f] generated on the fly (no 32KB z buffer)
    auto zval = [&](int m, int f) -> float {
        int e   = f & 63;
        int sel = f >> 6;                    // 0:Rh*o 1:Rh*M 2:|Rh-o| 3:|Rh-M|
        float rh    = Rh_s[m * RH_STRIDE + e];
        float other = (sel & 1) ? M_s[e] : o_s[e];
        return (sel < 2) ? (rh * other) : fabsf(rh - other);
    };

    for (int hop = 0; hop < 2; ++hop) {
        const size_t hop_base = ((size_t)b * 2 + hop);
        const float* Rbase = Rs + hop_base * (size_t)(NMEM * DIM * DIM);
        const float* tb    = ts + hop_base * (NMEM * DIM);

        // async-copy one 64x64 f32 R tile into padded LDS buffer `buf`
        auto issue_tile = [&](int m, int buf) {
            const float* gsrc = Rbase + (size_t)m * (DIM * DIM);
            #pragma unroll
            for (int i = 0; i < 4; ++i) {
                int c    = i * 256 + tid;        // float4-chunk id, 0..1023
                int drow = c >> 4;               // tile row (d)
                int col  = (c & 15) << 2;        // starting e within row
                const float* g = gsrc + c * 4;
                float*       l = &Rbuf[buf * TILE_F + drow * TILE_STRIDE + col];
                __builtin_amdgcn_global_load_async_to_lds_b128(
                    (gvec4i*)g, (lvec4i*)l, 0, 0);
            }
        };

        // ---- stage h[b,hop] (32x64), zero Rh accumulators, init logits ----
        {
            const float* hp = hs + hop_base * (NMEM * DIM);
            #pragma unroll
            for (int i = 0; i < 8; ++i) {
                int idx = tid + i * 256;
                h_s[idx] = hp[idx];
            }
        }
        #pragma unroll
        for (int i = tid; i < NMEM * RH_STRIDE; i += 256) Rh_s[i] = 0.0f;
        if (tid < NMEM) Z_s[tid] = W2_b[0];
        __builtin_prefetch(tb + tid * 32, 0, 0);     // warm the t tile (L2)
        __syncthreads();

        // ---- Rh[m][d] = sum_e R[m][d][e]*h[m][e]; double-buffered async ----
        issue_tile(0, 0);
        for (int m = 0; m < NMEM; ++m) {
            const int cur = m & 1;
            if (m < NMEM - 1) {
                issue_tile(m + 1, cur ^ 1);
                // oldest 4 async ops (= tile m) complete; async retires in order
                asm volatile("s_wait_asynccnt 4" ::: "memory");
            } else {
                asm volatile("s_wait_asynccnt 0" ::: "memory");
            }
            __syncthreads();                         // tile m visible to all

            const int d = tid & 63;
            const int q = tid >> 6;                  // quarter of the e-range
            const float* tile = &Rbuf[cur * TILE_F + d * TILE_STRIDE];
            const float* hm   = &h_s[m * DIM];
            float part = 0.0f;
            #pragma unroll
            for (int j = 0; j < 16; ++j) {
                int e = (q << 4) + j;
                part = fmaf(tile[e], hm[e], part);
            }
            atomicAdd(&Rh_s[m * RH_STRIDE + d], part);
            __syncthreads();                         // readers done before reuse
        }

        // ---- hcode = tanh(z @ W1^T + b1) via WMMA f32 16x16x4 ----
        v8f acc = {0.f, 0.f, 0.f, 0.f, 0.f, 0.f, 0.f, 0.f};
        #pragma unroll 4
        for (int k0 = 0; k0 < 256; k0 += 4) {
            int k = k0 + half * 2;
            v2f a, bb;
            a.x  = zval(mrow, k);
            a.y  = zval(mrow, k + 1);
            bb.x = W1row[k];
            bb.y = W1row[k + 1];
            acc = __builtin_amdgcn_wmma_f32_16x16x4_f32(
                false, a, false, bb, (short)0, acc, false, false);
        }

        // ---- logits: Z[m] += sum_d tanh(hcode)*W2[d] ----
        {
            float b1n = w1b_s[ncol];
            float w2n = w2_s[ncol];
            #pragma unroll
            for (int r = 0; r < 8; ++r) {
                int m = tile_m * 16 + r + half * 8;  // C/D layout: rows r / r+8
                float val = fast_tanhf(acc[r] + b1n) * w2n;
                #pragma unroll
                for (int off = 1; off < 16; off <<= 1)
                    val += __shfl_xor(val, off, 32); // 16 lanes share this m
                if (l16 == 0) atomicAdd(&Z_s[m], val);
            }
        }
        __syncthreads();

        // ---- softmax over 32 memories (wave 0) ----
        if (wave == 0) {
            float zv = Z_s[lane];
            float mx = zv;
            #pragma unroll
            for (int off = 16; off; off >>= 1)
                mx = fmaxf(mx, __shfl_xor(mx, off, 32));
            float ev = __expf(zv - mx);
            float sm = ev;
            #pragma unroll
            for (int off = 16; off; off >>= 1)
                sm += __shfl_xor(sm, off, 32);
            g_s[lane] = ev / sm;
        }
        __syncthreads();

        // ---- o_new[d] = sum_m t[m][d]*g[m] ----
        if (tid < DIM) {
            float on = 0.0f;
            #pragma unroll 4
            for (int m = 0; m < NMEM; ++m)
                on = fmaf(tb[m * DIM + tid], g_s[m], on);
            onew_s[tid] = on;
        }
        __syncthreads();

        // ---- M = relu([M, o_new, o] @ Wmem[hop]^T + bmem[hop]) ----
        float newM = 0.0f;
        if (tid < DIM) {
            const float* wm = Wmem_w + ((size_t)hop * DIM + tid) * 192;
            float a = Wmem_b[hop * DIM + tid];
            #pragma unroll 4
            for (int k = 0; k < DIM; ++k) a = fmaf(M_s[k],    wm[k],           a);
            #pragma unroll 4
            for (int k = 0; k < DIM; ++k) a = fmaf(onew_s[k], wm[DIM + k],     a);
            #pragma unroll 4
            for (int k = 0; k < DIM; ++k) a = fmaf(o_s[k],    wm[2 * DIM + k], a);
            newM = fmaxf(a, 0.0f);
        }
        __syncthreads();
        if (tid < DIM) {
            M_s[tid] = newM;
            o_s[tid] = onew_s[tid];
        }
        __syncthreads();
    }

    // ---- u = [M, vs] @ z_w^T + z_b ; out[b] = sum_d vs[d]*u[d] ----
    if (tid < DIM) {
        const float* zr = z_w + (size_t)tid * 128;
        const float* vb = vs + (size_t)b * DIM;
        float u = z_b[tid];
        #pragma unroll 4
        for (int k = 0; k < DIM; ++k) u = fmaf(M_s[k], zr[k],       u);
        #pragma unroll 4
        for (int k = 0; k < DIM; ++k) u = fmaf(vb[k],  zr[DIM + k], u);
        red_s[tid] = vb[tid] * u;
    }
    __syncthreads();
    if (wave == 0) {
        float v = red_s[lane] + red_s[lane + 32];
        #pragma unroll
        for (int off = 16; off; off >>= 1)
            v += __shfl_xor(v, off, 32);
        if (lane == 0) out[b] = v;
    }
}

extern "C" void kernel_launch(void* const* d_in, const int* in_sizes, int n_in,
                              void* d_out, int out_size, void* d_ws, size_t ws_size,
                              hipStream_t stream) {
    (void)in_sizes; (void)n_in; (void)d_ws; (void)ws_size; (void)out_size;
    const float* hs     = (const float*)d_in[0];
    const float* Rs     = (const float*)d_in[1];
    const float* ts     = (const float*)d_in[2];
    const float* vsp    = (const float*)d_in[3];
    const float* W1_w   = (const float*)d_in[4];
    const float* W1_b   = (const float*)d_in[5];
    const float* W2_w   = (const float*)d_in[6];
    const float* W2_b   = (const float*)d_in[7];
    const float* Wmem_w = (const float*)d_in[8];
    const float* Wmem_b = (const float*)d_in[9];
    const float* z_w    = (const float*)d_in[10];
    const float* z_b    = (const float*)d_in[11];
    float* outp = (float*)d_out;

    dim3 grid(1024), block(256);
    ripplenet_fwd_kernel<<<grid, block, 0, stream>>>(
        hs, Rs, ts, vsp, W1_w, W1_b, W2_w, W2_b, Wmem_w, Wmem_b, z_w, z_b, outp);
}